// decoderReLUNet_75419625717942
// MI455X (gfx1250) — compile-verified
//
#include <hip/hip_runtime.h>

// Closed-form reformulation of the ReLU-spline einsum as one dense GEMM:
//   OUT(1024x32) = ACT(1024x68) @ C(68x32), via V_WMMA_F32_16X16X4_F32.
// ACT[row][r] = relu(t[row] - r/64) for r<64;  t at r=64;  1 at r=65; 0 pad.
// C rows 0..63 are the partial-segment coeffs c[64m+r]; row 64 = alpha[m]
// (coefficient of t from fully-active segments); row 65 = beta[m] + init_val.
//
// Kernel 1 builds C pre-swizzled into the per-lane WMMA B-operand layout
// (Bpack[nt][kk][lane][2]) so the GEMM loads one b64 per K-step per lane.

typedef __attribute__((ext_vector_type(2))) float v2f;
typedef __attribute__((ext_vector_type(4))) float v4f;
typedef __attribute__((ext_vector_type(8))) float v8f;

#define RES_INV 0.015625f   // 1/64
#define NSTEP   17          // K = 68 / 4 WMMA steps
#define BPACK_ELEMS (2 * NSTEP * 32 * 2)   // [nt][kk][lane][j] = 2176 floats

// ---------------------------------------------------------------------------
// Kernel 1: segment sums A_q = sum_r c[64q+r], B_q = sum_r (r/64)c[64q+r];
// alpha[m] = sum_{q<m} A_q;  beta[m] = sum_{q<m}((m-q)A_q - B_q) + init_val.
// Then scatter logical C[r][m] into the WMMA B register layout:
//   Bpack[((nt*NSTEP + kk)*32 + lane)*2 + j] = C[4kk + 2*(lane>>4) + j][nt*16 + (lane&15)]
// (Knot j=2048 never fires since x = t+m < 32, so the c_eff tail is irrelevant.)
// ---------------------------------------------------------------------------
__global__ void __launch_bounds__(128)
build_bpack_kernel(const float* __restrict__ c, const float* __restrict__ init_val,
                   float* __restrict__ Bpack) {
    __shared__ float sA[32], sB[32];
    __shared__ float sAlpha[32], sBeta[32];
    const int tid = threadIdx.x;

    if (tid < 32) {
        float a = 0.0f, b = 0.0f;
        const float* seg = c + tid * 64;
        for (int r = 0; r < 64; ++r) {
            const float v = seg[r];
            a += v;
            b += v * ((float)r * RES_INV);
        }
        sA[tid] = a;
        sB[tid] = b;
    }
    __syncthreads();

    if (tid < 32) {
        const int m = tid;
        float alpha = 0.0f, beta = 0.0f;
        for (int q = 0; q < m; ++q) {
            alpha += sA[q];
            beta  += (float)(m - q) * sA[q] - sB[q];
        }
        sAlpha[m] = alpha;
        sBeta[m]  = beta + init_val[0];
    }
    __syncthreads();

    for (int idx = tid; idx < BPACK_ELEMS; idx += blockDim.x) {
        const int j    = idx & 1;
        const int lane = (idx >> 1) & 31;
        const int kk   = (idx >> 6) % NSTEP;
        const int nt   = (idx >> 6) / NSTEP;
        const int r    = (kk << 2) + ((lane >> 4) << 1) + j;
        const int m    = (nt << 4) + (lane & 15);
        float v;
        if      (r < 64)  v = c[m * 64 + r];
        else if (r == 64) v = sAlpha[m];
        else if (r == 65) v = sBeta[m];
        else              v = 0.0f;
        Bpack[idx] = v;
    }
}

// ---------------------------------------------------------------------------
// Kernel 2: one wave per 16x16 output tile (64 M-tiles x 2 N-tiles = 128 waves).
// A generated in registers; B is one b64 load per K-step from the pre-swizzled
// pack (per-lane address = base + lane*8, per-step immediate offset kk*256).
// ---------------------------------------------------------------------------
__device__ __forceinline__ float actval(float t, int r) {
    if (r < 64)  return fmaxf(t - (float)r * RES_INV, 0.0f);
    if (r == 64) return t;
    if (r == 65) return 1.0f;
    return 0.0f;
}

__global__ void __launch_bounds__(256)
phi_wmma_kernel(const float* __restrict__ tkh, const float* __restrict__ Bpack,
                float* __restrict__ out) {
    const int lane = threadIdx.x & 31;
    const int wave = (blockIdx.x * blockDim.x + threadIdx.x) >> 5;  // 0..127
    const int mt   = wave >> 1;            // M tile: 0..63
    const int nt   = wave & 1;             // N tile: 0..1
    const int row0 = mt << 4;              // first (b,k) row of this tile
    const int half = lane >> 4;            // 0: K={0,1}, 1: K={2,3} for A
    const int lan  = lane & 15;

    // A-matrix row value for this lane (16x4 f32 layout: M = lane%16).
    const float tv = tkh[row0 + lan];
    // Per-lane B pointer; step stride = 32 lanes * 2 floats = 32 v2f = 256 B.
    const v2f* bp = (const v2f*)(Bpack) + (nt * NSTEP * 32 + lane);

    v8f acc = {};
    #pragma unroll
    for (int kk = 0; kk < NSTEP; ++kk) {
        const int r0 = (kk << 2) + (half << 1);
        v2f a;
        a.x = actval(tv, r0);
        a.y = actval(tv, r0 + 1);
        const v2f b = bp[kk * 32];
        // 8 args: (neg_a, A, neg_b, B, c_mod, C, reuse_a, reuse_b)
        acc = __builtin_amdgcn_wmma_f32_16x16x4_f32(
            false, a, false, b, (short)0, acc, false, false);
    }

    // D layout: VGPR p holds M = p + 8*half, N = lan.  out[b][n][k] flat index
    // = b*128 + n*4 + k with bk = b*4+k, n = (32-m)%32.  The 8 accumulators are
    // two runs of 4 consecutive floats (k = 0..3) -> two b128 stores.
    const int col = (nt << 4) + lan;
    const int n   = (32 - col) & 31;
    const int b0  = (row0 + (half << 3)) >> 2;   // (b,k)-row group /4
    const v4f lo = {acc[0], acc[1], acc[2], acc[3]};
    const v4f hi = {acc[4], acc[5], acc[6], acc[7]};
    *(v4f*)(out + b0 * 128 + n * 4)       = lo;
    *(v4f*)(out + (b0 + 1) * 128 + n * 4) = hi;
}

extern "C" void kernel_launch(void* const* d_in, const int* in_sizes, int n_in,
                              void* d_out, int out_size, void* d_ws, size_t ws_size,
                              hipStream_t stream) {
    const float* t_k_hat  = (const float*)d_in[0];  // (256, 4)
    const float* c_relu   = (const float*)d_in[1];  // (2049,)
    // d_in[2] = shift, derived analytically (-j/64); unused.
    const float* init_val = (const float*)d_in[3];  // (1,)
    float*       out      = (float*)d_out;          // (256, 32, 4)
    float*       Bpack    = (float*)d_ws;           // 2176 floats = 8704 B

    build_bpack_kernel<<<1, 128, 0, stream>>>(c_relu, init_val, Bpack);
    // 128 tiles, 8 waves per 256-thread block -> 16 blocks.
    phi_wmma_kernel<<<16, 256, 0, stream>>>(t_k_hat, Bpack, out);
}